// EGNN_65876208386384
// MI455X (gfx1250) — compile-verified
//
#include <hip/hip_runtime.h>
#include <math.h>

// EGNN layer for MI455X (gfx1250): b=4, n=2048, d=64, m=16, K=32.
// All matrix math on V_WMMA_F32_16X16X4_F32 (full-f32 matrix path).
// Weights are pre-packed once into d_ws as zero-padded TRANSPOSED matrices so
// the WMMA inner loops are pure {ds_load_b64 A, global_load_b64 B, v_wmma}
// with no bounds guards / exec-mask branching.

typedef __attribute__((ext_vector_type(2))) float v2f;
typedef __attribute__((ext_vector_type(8))) float v8f;

#define Bb 4
#define Nn 2048
#define Dd 64
#define Mm 16
#define Kk 32

// Packed (transposed, padded) weight layout inside d_ws, in floats:
//   W1T [288][132]  : ew1[129,258] -> pad K 129->132, N 258->288
//   W2T [16][288]   : ew2[258,16]  -> pad K 258->288
//   CW1T[64][16]    : cw1[16,64]
//   NW1T[128][80]   : nw1[80,128]
//   NW2T[64][128]   : nw2[128,64]
#define OFF_W1T  0
#define OFF_W2T  (OFF_W1T + 288 * 132)      // 38016
#define OFF_CW1T (OFF_W2T + 16 * 288)       // 42624
#define OFF_NW1T (OFF_CW1T + 64 * 16)       // 43648
#define OFF_NW2T (OFF_NW1T + 128 * 80)      // 53888
#define PACK_TOTAL (OFF_NW2T + 64 * 128)    // 62080 (multiple of 64)
#define OFF_MI   PACK_TOTAL                 // m_i [B*N][16] after packed weights

__device__ __forceinline__ float silu_f(float x) {
    return x / (1.0f + __expf(-x));
}

// A fragment (16x4 f32) from LDS matrix with leading dim `ld` (floats).
// ISA 7.12.2: lanes 0-15 hold K={0,1} in v0/v1, lanes 16-31 hold K={2,3}.
__device__ __forceinline__ v2f ldsA(const float* Xs, int ld, int rowBase, int k0, int lane) {
    int m = lane & 15, kh = lane >> 4;
    return *(const v2f*)&Xs[(size_t)(rowBase + m) * ld + k0 + 2 * kh];
}

// B fragment (4x16 f32) from a TRANSPOSED padded weight Wt[NWp][KWp]:
// lane needs W[kr][n], W[kr+1][n] == Wt[n][kr], Wt[n][kr+1] -> one b64 load.
__device__ __forceinline__ v2f ldB_T(const float* __restrict__ Wt, int KWp,
                                     int k0, int n0, int lane) {
    int m = lane & 15, kh = lane >> 4;
    return *(const v2f*)&Wt[(size_t)(n0 + m) * KWp + k0 + 2 * kh];
}

#define WMMA_F32(a, b, c) \
    __builtin_amdgcn_wmma_f32_16x16x4_f32(false, (a), false, (b), (short)0, (c), false, false)

// ---------------------------------------------------------------------------
// Kernel 0: pack weights -> transposed + zero-padded layout in d_ws.
// ---------------------------------------------------------------------------
__global__ __launch_bounds__(256) void egnn_pack_kernel(
    const float* __restrict__ ew1, const float* __restrict__ ew2,
    const float* __restrict__ cw1, const float* __restrict__ nw1,
    const float* __restrict__ nw2, float* __restrict__ wp)
{
    int t = blockIdx.x * 256 + threadIdx.x;
    if (t >= PACK_TOTAL) return;
    float v;
    if (t < OFF_W2T) {                       // W1T [288][132]
        int n = t / 132, k = t % 132;
        v = (k < 129 && n < 258) ? ew1[(size_t)k * 258 + n] : 0.0f;
    } else if (t < OFF_CW1T) {               // W2T [16][288]
        int u = t - OFF_W2T;
        int n = u / 288, k = u % 288;
        v = (k < 258) ? ew2[(size_t)k * 16 + n] : 0.0f;
    } else if (t < OFF_NW1T) {               // CW1T [64][16]
        int u = t - OFF_CW1T;
        int n = u / 16, k = u % 16;
        v = cw1[(size_t)k * 64 + n];
    } else if (t < OFF_NW2T) {               // NW1T [128][80]
        int u = t - OFF_NW1T;
        int n = u / 80, k = u % 80;
        v = nw1[(size_t)k * 128 + n];
    } else {                                 // NW2T [64][128]
        int u = t - OFF_NW2T;
        int n = u / 128, k = u % 128;
        v = nw2[(size_t)k * 64 + n];
    }
    wp[t] = v;
}

// ---------------------------------------------------------------------------
// Kernel 1: per-node fused kNN + edge MLP + coors MLP.
// grid = b*n blocks, 256 threads (8 wave32).
// ---------------------------------------------------------------------------
__global__ __launch_bounds__(256) void egnn_edge_kernel(
    const float* __restrict__ feats, const float* __restrict__ coors,
    const float* __restrict__ wp,
    const float* __restrict__ eb1, const float* __restrict__ eb2,
    const float* __restrict__ cb1, const float* __restrict__ cb2,
    float* __restrict__ out_coors,  // [B,N,3]
    float* __restrict__ m_i)        // [B,N,16]
{
    const int tid  = threadIdx.x;
    const int lane = tid & 31;
    const int wave = tid >> 5;
    const int bi   = blockIdx.x >> 11;        // N = 2048
    const int i    = blockIdx.x & (Nn - 1);

    const float* W1T  = wp + OFF_W1T;
    const float* W2T  = wp + OFF_W2T;
    const float* CW1T = wp + OFF_CW1T;

    // Big reusable LDS arena (phases overlap in time, never concurrently):
    //  Phase A: coors row [2048*3] @0 (24576B) + dist [2048] @24576 (8192B)
    //  Phase B: X [32][132] @0 (16896B)        + H [32][288] @16896 (36864B)
    //  Phase C: Pp [4][32][16] @0 (8192B)  then T1 [32][64] @0 (8192B)
    __shared__ __align__(16) char smem[16896 + 36864];
    float* coorsS = (float*)smem;
    float* distS  = (float*)(smem + 24576);
    float* Xs     = (float*)smem;             // ld = 132
    float* Hs     = (float*)(smem + 16896);   // ld = 288
    float* Pp     = (float*)smem;             // GEMM2 partials [4][32][16]
    float* T1     = (float*)smem;             // ld = 64

    __shared__ int   kidx[Kk];
    __shared__ float kdist[Kk];
    __shared__ float krel[Kk * 3];
    __shared__ float M2s[Kk * Mm];            // m_ij [32][16]
    __shared__ float cwgt[Kk];
    __shared__ float redv[8];
    __shared__ int   redi[8];

    // ---- Phase A: distances + 32-NN selection (all in LDS) ----
    const float* crow = coors + (size_t)bi * Nn * 3;
    for (int t = tid; t < Nn * 3; t += 256) coorsS[t] = crow[t];
    __syncthreads();
    const float cx = coorsS[i * 3 + 0], cy = coorsS[i * 3 + 1], cz = coorsS[i * 3 + 2];
    for (int j = tid; j < Nn; j += 256) {
        float dx = cx - coorsS[j * 3 + 0];
        float dy = cy - coorsS[j * 3 + 1];
        float dz = cz - coorsS[j * 3 + 2];
        distS[j] = dx * dx + dy * dy + dz * dz;
    }
    __syncthreads();
    for (int r = 0; r < Kk; ++r) {
        float best = 3.0e38f; int bj = 0;
        for (int j = tid; j < Nn; j += 256) {
            float v = distS[j];
            if (v < best) { best = v; bj = j; }
        }
        // wave32 min-reduce via lane shuffles
        for (int s = 16; s > 0; s >>= 1) {
            float ov = __shfl_xor(best, s, 32);
            int   oi = __shfl_xor(bj,   s, 32);
            if (ov < best) { best = ov; bj = oi; }
        }
        if (lane == 0) { redv[wave] = best; redi[wave] = bj; }
        __syncthreads();
        if (tid == 0) {
            float bv = redv[0]; int w = redi[0];
            for (int q = 1; q < 8; ++q)
                if (redv[q] < bv) { bv = redv[q]; w = redi[q]; }
            kidx[r]  = w;
            kdist[r] = bv;
            distS[w] = 3.0e38f;
            krel[r * 3 + 0] = cx - coorsS[w * 3 + 0];
            krel[r * 3 + 1] = cy - coorsS[w * 3 + 1];
            krel[r * 3 + 2] = cz - coorsS[w * 3 + 2];
        }
        __syncthreads();
    }

    // ---- Phase B: build edge input X[32][132] = [f_i(64) | f_j(64) | d2 | pad] ----
    const float* fi = feats + ((size_t)bi * Nn + i) * Dd;
    for (int t = tid; t < Kk * 132; t += 256) {
        int k = t / 132, c = t % 132;
        float v;
        if (c < 64)        v = fi[c];
        else if (c < 128)  v = feats[((size_t)bi * Nn + kidx[k]) * Dd + (c - 64)];
        else if (c == 128) v = kdist[k];
        else               v = 0.0f;
        Xs[t] = v;
    }
    __syncthreads();

    // GEMM1: X[32,132] @ W1[132,288] -> H[32,288]; 2 M-tiles x 18 N-tiles.
    for (int job = wave; job < 36; job += 8) {
        int mt = job / 18, nt = job % 18;
        int rowBase = mt * 16, n0 = nt * 16;
        __builtin_prefetch(&W1T[(size_t)(n0 + (lane & 15)) * 132], 0, 1);
        v8f acc = {};
        for (int ks = 0; ks < 33; ++ks) {
            int k0 = ks * 4;
            v2f a = ldsA(Xs, 132, rowBase, k0, lane);
            v2f b = ldB_T(W1T, 132, k0, n0, lane);
            acc = WMMA_F32(a, b, acc);
        }
        int m = lane & 15, kh = lane >> 4;
        int col = n0 + m;
        float bias = (col < 258) ? eb1[col] : 0.0f;
        union { v8f v; float f[8]; } u; u.v = acc;
        for (int r = 0; r < 8; ++r) {
            int row = rowBase + r + 8 * kh;
            Hs[row * 288 + col] = silu_f(u.f[r] + bias);  // padded cols -> silu(0)=0
        }
    }
    __syncthreads();

    // GEMM2: H[32,288] @ W2[288,16] -> m_ij[32,16].
    // Split-K: 8 jobs = 2 M-tiles x 4 K-chunks (18 k-steps each), partials in LDS.
    {
        int mt = wave & 1, kc = wave >> 1;
        int rowBase = mt * 16;
        v8f acc = {};
        for (int ks = kc * 18; ks < kc * 18 + 18; ++ks) {
            int k0 = ks * 4;
            v2f a = ldsA(Hs, 288, rowBase, k0, lane);
            v2f b = ldB_T(W2T, 288, k0, 0, lane);
            acc = WMMA_F32(a, b, acc);
        }
        int m = lane & 15, kh = lane >> 4;
        union { v8f v; float f[8]; } u; u.v = acc;
        for (int r = 0; r < 8; ++r) {
            int row = rowBase + r + 8 * kh;
            Pp[(kc * 32 + row) * Mm + m] = u.f[r];
        }
    }
    __syncthreads();
    for (int t = tid; t < Kk * Mm; t += 256) {
        int row = t / Mm, c = t % Mm;
        float s = Pp[t] + Pp[(32 + row) * Mm + c]
                + Pp[(64 + row) * Mm + c] + Pp[(96 + row) * Mm + c];
        M2s[t] = silu_f(s + eb2[c]);
    }
    __syncthreads();

    // GEMM3: m_ij[32,16] @ cw1[16,64] -> T1[32,64]; exactly 8 jobs, 1 per wave.
    {
        int mt = wave >> 2, nt = wave & 3;
        int rowBase = mt * 16, n0 = nt * 16;
        v8f acc = {};
        for (int ks = 0; ks < 4; ++ks) {
            v2f a = ldsA(M2s, Mm, rowBase, ks * 4, lane);
            v2f b = ldB_T(CW1T, 16, ks * 4, n0, lane);
            acc = WMMA_F32(a, b, acc);
        }
        int m = lane & 15, kh = lane >> 4;
        int col = n0 + m;
        float bias = cb1[col];
        union { v8f v; float f[8]; } u; u.v = acc;
        for (int r = 0; r < 8; ++r) {
            int row = rowBase + r + 8 * kh;
            T1[row * 64 + col] = silu_f(u.f[r] + bias);
        }
    }
    __syncthreads();

    // coor_w[e] = T1[e,:] . cw2 + cb2  (cw2 passed via cb2-adjacent pointers below)
    if (tid < Kk) {
        float s = cb2[64];                    // cb2 scalar stored at cw2[64]
        for (int c = 0; c < 64; ++c) s += T1[tid * 64 + c] * cb2[c];
        cwgt[tid] = s;
    }
    __syncthreads();
    // coors_out = coors + sum_k coor_w[k] * rel_k
    if (tid < 3) {
        float s = coors[((size_t)bi * Nn + i) * 3 + tid];
        for (int e = 0; e < Kk; ++e) s += cwgt[e] * krel[e * 3 + tid];
        out_coors[((size_t)bi * Nn + i) * 3 + tid] = s;
    }
    // m_i = sum over neighbors of m_ij
    if (tid >= 32 && tid < 32 + Mm) {
        int c = tid - 32;
        float s = 0.0f;
        for (int e = 0; e < Kk; ++e) s += M2s[e * Mm + c];
        m_i[((size_t)bi * Nn + i) * Mm + c] = s;
    }
}

// Small helper kernel to pack [cw2(64) | cb2(1)] contiguously in ws (after mi).
__global__ void egnn_pack_cv_kernel(const float* __restrict__ cw2,
                                    const float* __restrict__ cb2,
                                    float* __restrict__ dst)
{
    int t = threadIdx.x;
    if (t < 64) dst[t] = cw2[t];
    if (t == 64) dst[64] = cb2[0];
}

// ---------------------------------------------------------------------------
// Kernel 2: node MLP with residual. grid = (b*n)/32 blocks, 256 threads.
// ---------------------------------------------------------------------------
__global__ __launch_bounds__(256) void egnn_node_kernel(
    const float* __restrict__ feats, const float* __restrict__ m_i,
    const float* __restrict__ wp,
    const float* __restrict__ nb1, const float* __restrict__ nb2,
    float* __restrict__ out_nodes)  // [B,N,64]
{
    const int tid  = threadIdx.x;
    const int lane = tid & 31;
    const int wave = tid >> 5;
    const size_t row0 = (size_t)blockIdx.x * 32;  // row in flattened [B*N]

    const float* NW1T = wp + OFF_NW1T;
    const float* NW2T = wp + OFF_NW2T;

    __shared__ __align__(16) float Xn[32 * 80];   // [feats(64) | m_i(16)]
    __shared__ __align__(16) float H1[32 * 128];

    for (int t = tid; t < 32 * 80; t += 256) {
        int r = t / 80, c = t % 80;
        size_t gr = row0 + r;
        Xn[t] = (c < 64) ? feats[gr * 64 + c] : m_i[gr * 16 + (c - 64)];
    }
    __syncthreads();

    // GEMM1: Xn[32,80] @ nw1[80,128] -> H1; 16 jobs over 8 waves.
    for (int job = wave; job < 16; job += 8) {
        int mt = job >> 3, nt = job & 7;
        int rowBase = mt * 16, n0 = nt * 16;
        v8f acc = {};
        for (int ks = 0; ks < 20; ++ks) {
            v2f a = ldsA(Xn, 80, rowBase, ks * 4, lane);
            v2f b = ldB_T(NW1T, 80, ks * 4, n0, lane);
            acc = WMMA_F32(a, b, acc);
        }
        int m = lane & 15, kh = lane >> 4;
        int col = n0 + m;
        float bias = nb1[col];
        union { v8f v; float f[8]; } u; u.v = acc;
        for (int r = 0; r < 8; ++r) {
            int row = rowBase + r + 8 * kh;
            H1[row * 128 + col] = silu_f(u.f[r] + bias);
        }
    }
    __syncthreads();

    // GEMM2: H1[32,128] @ nw2[128,64] -> out (+bias +residual); 8 jobs, 1/wave.
    {
        int mt = wave >> 2, nt = wave & 3;
        int rowBase = mt * 16, n0 = nt * 16;
        v8f acc = {};
        for (int ks = 0; ks < 32; ++ks) {
            v2f a = ldsA(H1, 128, rowBase, ks * 4, lane);
            v2f b = ldB_T(NW2T, 128, ks * 4, n0, lane);
            acc = WMMA_F32(a, b, acc);
        }
        int m = lane & 15, kh = lane >> 4;
        int col = n0 + m;
        float bias = nb2[col];
        union { v8f v; float f[8]; } u; u.v = acc;
        for (int r = 0; r < 8; ++r) {
            int row = rowBase + r + 8 * kh;
            size_t gr = row0 + row;
            out_nodes[gr * 64 + col] = u.f[r] + bias + feats[gr * 64 + col];
        }
    }
}

extern "C" void kernel_launch(void* const* d_in, const int* in_sizes, int n_in,
                              void* d_out, int out_size, void* d_ws, size_t ws_size,
                              hipStream_t stream) {
    (void)in_sizes; (void)n_in; (void)out_size; (void)ws_size;
    const float* feats = (const float*)d_in[0];
    const float* coors = (const float*)d_in[1];
    const float* ew1   = (const float*)d_in[2];
    const float* eb1   = (const float*)d_in[3];
    const float* ew2   = (const float*)d_in[4];
    const float* eb2   = (const float*)d_in[5];
    const float* cw1   = (const float*)d_in[6];
    const float* cb1   = (const float*)d_in[7];
    const float* cw2   = (const float*)d_in[8];
    const float* cb2   = (const float*)d_in[9];
    const float* nw1   = (const float*)d_in[10];
    const float* nb1   = (const float*)d_in[11];
    const float* nw2   = (const float*)d_in[12];
    const float* nb2   = (const float*)d_in[13];

    float* wsf = (float*)d_ws;
    float* wp  = wsf;                                  // packed weights
    float* mi  = wsf + OFF_MI;                         // m_i [B*N][16]
    float* cv  = mi + (size_t)Bb * Nn * Mm;            // [cw2(64) | cb2(1)]

    float* out_nodes = (float*)d_out;                  // [4,2048,64]
    float* out_coors = out_nodes + (size_t)Bb * Nn * Dd; // [4,2048,3]

    egnn_pack_kernel<<<(PACK_TOTAL + 255) / 256, 256, 0, stream>>>(
        ew1, ew2, cw1, nw1, nw2, wp);
    egnn_pack_cv_kernel<<<1, 128, 0, stream>>>(cw2, cb2, cv);
    egnn_edge_kernel<<<Bb * Nn, 256, 0, stream>>>(
        feats, coors, wp, eb1, eb2, cb1, cv, out_coors, mi);
    egnn_node_kernel<<<(Bb * Nn) / 32, 256, 0, stream>>>(
        feats, mi, wp, nb1, nb2, out_nodes);
}